// GIN_68367289418045
// MI455X (gfx1250) — compile-verified
//
#include <hip/hip_runtime.h>
#include <hip/hip_bf16.h>

typedef __attribute__((ext_vector_type(2))) float v2f;
typedef __attribute__((ext_vector_type(8))) float v8f;

#define HID   52
#define HID2  104
#define ZH_LD 112   // zh row stride padded to 7*16 so full WMMA tiles store in-bounds

// ---------------------------------------------------------------------------
// Edge aggregation: agg[dst] += h[src].  One thread per (edge, float4-chunk).
// 52 floats = 13 float4 per edge; consecutive threads -> consecutive chunks of
// the same edge row => coalesced 208B gathers. Atomics land in L2 (entire
// node-feature matrix is 20.8MB, L2-resident in MI455X's 192MB L2).
// ---------------------------------------------------------------------------
__global__ __launch_bounds__(256)
void gin_scatter_kernel(const float* __restrict__ h,
                        const int* __restrict__ src,
                        const int* __restrict__ dst,
                        float* __restrict__ agg, int n_edges) {
    int t = blockIdx.x * blockDim.x + threadIdx.x;
    int e  = t / 13;
    int f4 = (t % 13) * 4;
    if (e >= n_edges) return;
    int s = src[e];
    int d = dst[e];
    const float4 v = *(const float4*)(h + (size_t)s * HID + f4);
    float* base = agg + (size_t)d * HID + f4;
    atomicAdd(base + 0, v.x);
    atomicAdd(base + 1, v.y);
    atomicAdd(base + 2, v.z);
    atomicAdd(base + 3, v.w);
}

// ---------------------------------------------------------------------------
// GEMM1: zh = relu(((1+eps)*h + agg) @ W1 + b1)   [N,52]x[52,104] -> [N,112pad]
// One wave computes a full 16x112 strip: A fragment loaded once per k-step,
// reused across 7 independent WMMA accumulator chains (ILP across the XDL).
// Column indices are CLAMPED, not guarded: junk from clamped lanes only lands
// in zh columns >= 104, which GEMM2 never reads. No EXEC divergence anywhere.
// ---------------------------------------------------------------------------
__global__ __launch_bounds__(256)
void gin_gemm1_kernel(const float* __restrict__ h,
                      const float* __restrict__ agg,
                      const float* __restrict__ w1,
                      const float* __restrict__ b1,
                      const float* __restrict__ eps, int layer,
                      float* __restrict__ zh, int n_nodes) {
    const int NT = 7;
    int mtiles = n_nodes >> 4;
    int mt = blockIdx.x * (blockDim.x >> 5) + (threadIdx.x >> 5);
    if (mt >= mtiles) return;                   // wave-uniform: EXEC stays full
    int lane = threadIdx.x & 31;
    int half = lane >> 4;                       // K pair select within frag
    int lidx = lane & 15;

    float epsv = 1.0f + eps[layer];
    const float* W  = w1 + (size_t)layer * HID * HID2;
    const float* B1 = b1 + (size_t)layer * HID2;

    const float* hp = h   + (size_t)(mt * 16 + lidx) * HID + (half << 1);
    const float* gp = agg + (size_t)(mt * 16 + lidx) * HID + (half << 1);

    int nc[NT];
    const float* wp[NT];
#pragma unroll
    for (int t = 0; t < NT; ++t) {
        int n = t * 16 + lidx;
        nc[t] = (n < HID2) ? n : (HID2 - 1);    // clamp instead of guard
        wp[t] = W + (size_t)(half << 1) * HID2 + nc[t];
    }

    v8f c[NT] = {};
#pragma unroll
    for (int k0 = 0; k0 < HID; k0 += 4) {
        v2f hv = *(const v2f*)(hp + k0);
        v2f gv = *(const v2f*)(gp + k0);
        v2f a;
        a.x = epsv * hv.x + gv.x;               // fused (1+eps)*h + agg
        a.y = epsv * hv.y + gv.y;
#pragma unroll
        for (int t = 0; t < NT; ++t) {
            v2f b;
            b.x = wp[t][(size_t)k0 * HID2];     // constant imm offsets
            b.y = wp[t][(size_t)(k0 + 1) * HID2];
            c[t] = __builtin_amdgcn_wmma_f32_16x16x4_f32(false, a, false, b,
                                                         (short)0, c[t], false, false);
        }
    }
#pragma unroll
    for (int t = 0; t < NT; ++t) {
        float bias = B1[nc[t]];
#pragma unroll
        for (int r = 0; r < 8; ++r) {
            int row = mt * 16 + r + half * 8;
            float v = c[t][r] + bias;
            zh[(size_t)row * ZH_LD + t * 16 + lidx] = v > 0.0f ? v : 0.0f;
        }
    }
}

// ---------------------------------------------------------------------------
// GEMM2: hout = relu(zh @ W2 + b2)   [N,104]x[104,52] -> [N,52]
// One wave per 16x64 strip (4 tiles, stores guarded to 52 cols), K=104.
// ---------------------------------------------------------------------------
__global__ __launch_bounds__(256)
void gin_gemm2_kernel(const float* __restrict__ zh,
                      const float* __restrict__ w2,
                      const float* __restrict__ b2, int layer,
                      float* __restrict__ hout, int n_nodes) {
    const int NT = 4;
    int mtiles = n_nodes >> 4;
    int mt = blockIdx.x * (blockDim.x >> 5) + (threadIdx.x >> 5);
    if (mt >= mtiles) return;
    int lane = threadIdx.x & 31;
    int half = lane >> 4;
    int lidx = lane & 15;

    const float* W  = w2 + (size_t)layer * HID2 * HID;
    const float* B2 = b2 + (size_t)layer * HID;

    const float* ap = zh + (size_t)(mt * 16 + lidx) * ZH_LD + (half << 1);

    int nc[NT];
    const float* wp[NT];
#pragma unroll
    for (int t = 0; t < NT; ++t) {
        int n = t * 16 + lidx;
        nc[t] = (n < HID) ? n : (HID - 1);
        wp[t] = W + (size_t)(half << 1) * HID + nc[t];
    }

    v8f c[NT] = {};
#pragma unroll
    for (int k0 = 0; k0 < HID2; k0 += 4) {
        v2f a = *(const v2f*)(ap + k0);
#pragma unroll
        for (int t = 0; t < NT; ++t) {
            v2f b;
            b.x = wp[t][(size_t)k0 * HID];
            b.y = wp[t][(size_t)(k0 + 1) * HID];
            c[t] = __builtin_amdgcn_wmma_f32_16x16x4_f32(false, a, false, b,
                                                         (short)0, c[t], false, false);
        }
    }
#pragma unroll
    for (int t = 0; t < NT; ++t) {
        float bias = B2[nc[t]];
        if (t * 16 + lidx < HID) {              // only tile 3 partially masked
#pragma unroll
            for (int r = 0; r < 8; ++r) {
                int row = mt * 16 + r + half * 8;
                float v = c[t][r] + bias;
                hout[(size_t)row * HID + t * 16 + lidx] = v > 0.0f ? v : 0.0f;
            }
        }
    }
}

// ---------------------------------------------------------------------------
// Final linear, accumulated per stage instead of materializing the concat:
//   init=1:  out = lin_b + h @ lin_w[52*block : ...]
//   init=0:  out +=         h @ lin_w[52*block : ...]
// C fragments seeded from bias / previous out; WMMA chains accumulate on top.
// Seed reads use clamped columns (in-bounds junk, discarded by guarded store).
// ---------------------------------------------------------------------------
__global__ __launch_bounds__(256)
void gin_lin_kernel(const float* __restrict__ h,
                    const float* __restrict__ lin_w,
                    const float* __restrict__ lin_b,
                    float* __restrict__ out, int n_nodes, int block, int init) {
    const int NT = 4;
    int mtiles = n_nodes >> 4;
    int mt = blockIdx.x * (blockDim.x >> 5) + (threadIdx.x >> 5);
    if (mt >= mtiles) return;
    int lane = threadIdx.x & 31;
    int half = lane >> 4;
    int lidx = lane & 15;

    const float* W = lin_w + (size_t)block * HID * HID;  // rows 52*block.. of [208,52]
    const float* ap = h + (size_t)(mt * 16 + lidx) * HID + (half << 1);

    int nc[NT];
    const float* wp[NT];
#pragma unroll
    for (int t = 0; t < NT; ++t) {
        int n = t * 16 + lidx;
        nc[t] = (n < HID) ? n : (HID - 1);
        wp[t] = W + (size_t)(half << 1) * HID + nc[t];
    }

    v8f c[NT];
    if (init) {                                  // uniform branch
#pragma unroll
        for (int t = 0; t < NT; ++t) {
            float bias = lin_b[nc[t]];
#pragma unroll
            for (int r = 0; r < 8; ++r) c[t][r] = bias;
        }
    } else {
#pragma unroll
        for (int t = 0; t < NT; ++t) {
#pragma unroll
            for (int r = 0; r < 8; ++r) {
                int row = mt * 16 + r + half * 8;
                c[t][r] = out[(size_t)row * HID + nc[t]];   // clamped: in-bounds
            }
        }
    }
#pragma unroll
    for (int k0 = 0; k0 < HID; k0 += 4) {
        v2f a = *(const v2f*)(ap + k0);
#pragma unroll
        for (int t = 0; t < NT; ++t) {
            v2f b;
            b.x = wp[t][(size_t)k0 * HID];
            b.y = wp[t][(size_t)(k0 + 1) * HID];
            c[t] = __builtin_amdgcn_wmma_f32_16x16x4_f32(false, a, false, b,
                                                         (short)0, c[t], false, false);
        }
    }
#pragma unroll
    for (int t = 0; t < NT; ++t) {
        if (t * 16 + lidx < HID) {
#pragma unroll
            for (int r = 0; r < 8; ++r) {
                int row = mt * 16 + r + half * 8;
                out[(size_t)row * HID + t * 16 + lidx] = c[t][r];
            }
        }
    }
}

// ---------------------------------------------------------------------------
extern "C" void kernel_launch(void* const* d_in, const int* in_sizes, int n_in,
                              void* d_out, int out_size, void* d_ws, size_t ws_size,
                              hipStream_t stream) {
    const float* x     = (const float*)d_in[0];
    const int*   ei    = (const int*)  d_in[1];
    const float* w1    = (const float*)d_in[2];
    const float* b1    = (const float*)d_in[3];
    const float* w2    = (const float*)d_in[4];
    const float* b2    = (const float*)d_in[5];
    const float* eps   = (const float*)d_in[6];
    const float* lin_w = (const float*)d_in[7];
    const float* lin_b = (const float*)d_in[8];
    float* out = (float*)d_out;

    const int n_nodes = in_sizes[0] / HID;      // 100000 (multiple of 16)
    const int n_edges = in_sizes[1] / 2;        // 3200000
    const int* src = ei;
    const int* dst = ei + n_edges;

    // Workspace carve-up (all 16B-aligned): h_buf | agg | zh(padded to 112)
    float* h_buf = (float*)d_ws;
    float* agg   = h_buf + (size_t)n_nodes * HID;
    float* zh    = agg   + (size_t)n_nodes * HID;

    const int threads = 256;
    const int waves_per_blk = threads / 32;
    const int mtiles = n_nodes / 16;            // 6250 row-strips, 1 wave each
    const int g_blocks = (mtiles + waves_per_blk - 1) / waves_per_blk;
    const int sc_threads = n_edges * 13;
    const int sc_blocks  = (sc_threads + threads - 1) / threads;

    // out = lin_b + x @ lin_w[0:52,:]
    gin_lin_kernel<<<g_blocks, threads, 0, stream>>>(x, lin_w, lin_b, out,
                                                     n_nodes, 0, 1);

    const float* h = x;
    for (int l = 0; l < 3; ++l) {
        hipMemsetAsync(agg, 0, (size_t)n_nodes * HID * sizeof(float), stream);
        gin_scatter_kernel<<<sc_blocks, threads, 0, stream>>>(h, src, dst, agg, n_edges);
        gin_gemm1_kernel<<<g_blocks, threads, 0, stream>>>(h, agg, w1, b1, eps, l,
                                                           zh, n_nodes);
        gin_gemm2_kernel<<<g_blocks, threads, 0, stream>>>(zh, w2, b2, l,
                                                           h_buf, n_nodes);
        gin_lin_kernel<<<g_blocks, threads, 0, stream>>>(h_buf, lin_w, lin_b, out,
                                                         n_nodes, l + 1, 0);
        h = h_buf;
    }
}